// LoRALinear_52012053954937
// MI455X (gfx1250) — compile-verified
//
#include <hip/hip_runtime.h>

typedef float v2f __attribute__((ext_vector_type(2)));
typedef float v4f __attribute__((ext_vector_type(4)));
typedef float v8f __attribute__((ext_vector_type(8)));

#define M_TOTAL (8 * 4096)
#define KDIM 1024
#define NDIM 1024
#define RANK 16
#define SCALING 1.0f

#define BM 128
#define BN 128
#define BK 16
#define XS 20   // padded LDS row stride in floats: 80B = 16B-aligned, bank-conflict friendly
#define NIT (KDIM / BK)   // 64 K-tiles

__global__ __launch_bounds__(256, 2)
void lora_linear_wmma_f32(const float* __restrict__ x,
                          const float* __restrict__ W,
                          const float* __restrict__ bias,
                          const float* __restrict__ lora_A,
                          const float* __restrict__ lora_B,
                          float* __restrict__ out)
{
    __shared__ float xls[2][BM * XS];    // double-buffered x tile   [128][BK]
    __shared__ float wls[2][BN * XS];    // double-buffered W tile   [128][BK]
    __shared__ float als[2][RANK * XS];  // double-buffered A tile   [16][BK]
    __shared__ float xabuf[BM * XS];     // scaled xa = x@A^T        [128][16]

    const int tid    = threadIdx.x;
    const int lane   = tid & 31;
    const int wid    = tid >> 5;     // 8 waves
    const int wave_m = wid >> 1;     // 0..3 -> 32-row group
    const int wave_n = wid & 1;      // 0..1 -> 64-col group

    // wave-uniform predicate via readfirstlane -> scalar branch, no EXEC masking
    const bool lora_wave = (__builtin_amdgcn_readfirstlane(wave_n) == 0);

    const int m_base = (int)blockIdx.x * BM;
    const int n_base = (int)blockIdx.y * BN;

    const int mr = lane & 15;        // row (A) / col (B) index within 16-tile
    const int kh = lane >> 4;        // 0: K-pair {0,1}; 1: K-pair {2,3}

    // global->LDS staging geometry: each thread moves 2 float4 per 128-row tile
    const int lrow  = tid >> 2;          // 0..63
    const int lquad = (tid & 3) << 2;    // 0,4,8,12
    const int ar    = tid >> 2;          // 0..15 for A tile (threads 0..63)
    const int aq    = lquad;

    v8f acc[2][4];
    v8f xa[2];
#pragma unroll
    for (int i = 0; i < 2; ++i) {
        xa[i] = (v8f)0.0f;
#pragma unroll
        for (int j = 0; j < 4; ++j) acc[i][j] = (v8f)0.0f;
    }

    const float* gx = x + (size_t)(m_base + lrow) * KDIM + lquad;
    const float* gw = W + (size_t)(n_base + lrow) * KDIM + lquad;
    const float* ga = lora_A + (size_t)ar * KDIM + aq;

    // ---- prologue: stage K-tile 0 into registers ----
    v4f rx0 = *(const v4f*)(gx);
    v4f rx1 = *(const v4f*)(gx + (size_t)64 * KDIM);
    v4f rw0 = *(const v4f*)(gw);
    v4f rw1 = *(const v4f*)(gw + (size_t)64 * KDIM);
    v4f ra  = (v4f)0.0f;
    if (tid < 64) ra = *(const v4f*)(ga);

    int p = 0;
    for (int it = 0; it < NIT; ++it) {
        // (a) commit staged registers to LDS buffer p
        *(v4f*)&xls[p][lrow * XS + lquad]        = rx0;
        *(v4f*)&xls[p][(lrow + 64) * XS + lquad] = rx1;
        *(v4f*)&wls[p][lrow * XS + lquad]        = rw0;
        *(v4f*)&wls[p][(lrow + 64) * XS + lquad] = rw1;
        if (tid < 64) *(v4f*)&als[p][ar * XS + aq] = ra;

        // (b) issue next tile's global loads (consumed next iteration)
        if (it + 1 < NIT) {
            const int k0 = (it + 1) * BK;
            rx0 = *(const v4f*)(gx + k0);
            rx1 = *(const v4f*)(gx + (size_t)64 * KDIM + k0);
            rw0 = *(const v4f*)(gw + k0);
            rw1 = *(const v4f*)(gw + (size_t)64 * KDIM + k0);
            if (tid < 64) ra = *(const v4f*)(ga + k0);
        }

        // (c) single barrier per K-tile
        __syncthreads();

        // (d) 4 k-steps of WMMA out of buffer p
        v2f amf[2][4];
#pragma unroll
        for (int ks = 0; ks < 4; ++ks) {
            const int kk = ks * 4 + 2 * kh;  // this lane's K pair within the tile
            amf[0][ks] = *(const v2f*)&xls[p][(wave_m * 32 + mr) * XS + kk];
            amf[1][ks] = *(const v2f*)&xls[p][(wave_m * 32 + 16 + mr) * XS + kk];
            v2f bn0 = *(const v2f*)&wls[p][(wave_n * 64 + mr) * XS + kk];
            v2f bn1 = *(const v2f*)&wls[p][(wave_n * 64 + 16 + mr) * XS + kk];
            v2f bn2 = *(const v2f*)&wls[p][(wave_n * 64 + 32 + mr) * XS + kk];
            v2f bn3 = *(const v2f*)&wls[p][(wave_n * 64 + 48 + mr) * XS + kk];

            acc[0][0] = __builtin_amdgcn_wmma_f32_16x16x4_f32(false, amf[0][ks], false, bn0, (short)0, acc[0][0], false, false);
            acc[0][1] = __builtin_amdgcn_wmma_f32_16x16x4_f32(false, amf[0][ks], false, bn1, (short)0, acc[0][1], false, false);
            acc[0][2] = __builtin_amdgcn_wmma_f32_16x16x4_f32(false, amf[0][ks], false, bn2, (short)0, acc[0][2], false, false);
            acc[0][3] = __builtin_amdgcn_wmma_f32_16x16x4_f32(false, amf[0][ks], false, bn3, (short)0, acc[0][3], false, false);
            acc[1][0] = __builtin_amdgcn_wmma_f32_16x16x4_f32(false, amf[1][ks], false, bn0, (short)0, acc[1][0], false, false);
            acc[1][1] = __builtin_amdgcn_wmma_f32_16x16x4_f32(false, amf[1][ks], false, bn1, (short)0, acc[1][1], false, false);
            acc[1][2] = __builtin_amdgcn_wmma_f32_16x16x4_f32(false, amf[1][ks], false, bn2, (short)0, acc[1][2], false, false);
            acc[1][3] = __builtin_amdgcn_wmma_f32_16x16x4_f32(false, amf[1][ks], false, bn3, (short)0, acc[1][3], false, false);
        }

        if (lora_wave) {  // scalar-uniform branch: EXEC stays all-ones for WMMA
#pragma unroll
            for (int ks = 0; ks < 4; ++ks) {
                const int kk = ks * 4 + 2 * kh;
                v2f ba = *(const v2f*)&als[p][mr * XS + kk];
                xa[0] = __builtin_amdgcn_wmma_f32_16x16x4_f32(false, amf[0][ks], false, ba, (short)0, xa[0], false, false);
                xa[1] = __builtin_amdgcn_wmma_f32_16x16x4_f32(false, amf[1][ks], false, ba, (short)0, xa[1], false, false);
            }
        }

        p ^= 1;
    }

    // ---- epilogue: y += SCALING * xa @ lora_B^T, + bias ----
    __syncthreads();
    if (lora_wave) {
        // spill xa (C/D layout) to LDS, scaled: VGPR v holds row v (+8 for upper half-wave)
#pragma unroll
        for (int i = 0; i < 2; ++i) {
            const int rbase = wave_m * 32 + i * 16 + 8 * kh;
#pragma unroll
            for (int v = 0; v < 8; ++v)
                xabuf[(rbase + v) * XS + mr] = xa[i][v] * SCALING;
        }
    }
    __syncthreads();

#pragma unroll
    for (int ks = 0; ks < 4; ++ks) {
        const int kk = ks * 4 + 2 * kh;  // rank index pair for this lane
        v2f am0 = *(const v2f*)&xabuf[(wave_m * 32 + mr) * XS + kk];
        v2f am1 = *(const v2f*)&xabuf[(wave_m * 32 + 16 + mr) * XS + kk];
#pragma unroll
        for (int j = 0; j < 4; ++j) {
            const int ng = n_base + wave_n * 64 + j * 16 + mr;
            v2f bb = *(const v2f*)(lora_B + (size_t)ng * RANK + kk);  // B_op[r][n] = lora_B[n][r]
            acc[0][j] = __builtin_amdgcn_wmma_f32_16x16x4_f32(false, am0, false, bb, (short)0, acc[0][j], false, false);
            acc[1][j] = __builtin_amdgcn_wmma_f32_16x16x4_f32(false, am1, false, bb, (short)0, acc[1][j], false, false);
        }
    }

    // store with fused bias
#pragma unroll
    for (int i = 0; i < 2; ++i) {
        const int mg = m_base + wave_m * 32 + i * 16 + 8 * kh;
#pragma unroll
        for (int j = 0; j < 4; ++j) {
            const int ng = n_base + wave_n * 64 + j * 16 + mr;
            const float bv = bias[ng];
#pragma unroll
            for (int v = 0; v < 8; ++v)
                out[(size_t)(mg + v) * NDIM + ng] = acc[i][j][v] + bv;
        }
    }
}

extern "C" void kernel_launch(void* const* d_in, const int* in_sizes, int n_in,
                              void* d_out, int out_size, void* d_ws, size_t ws_size,
                              hipStream_t stream) {
    (void)in_sizes; (void)n_in; (void)d_ws; (void)ws_size; (void)out_size;
    const float* x      = (const float*)d_in[0];
    const float* W      = (const float*)d_in[1];
    const float* b      = (const float*)d_in[2];
    const float* lora_A = (const float*)d_in[3];
    const float* lora_B = (const float*)d_in[4];
    float* out = (float*)d_out;

    dim3 grid(M_TOTAL / BM, NDIM / BN);  // 256 x 8 workgroups
    lora_linear_wmma_f32<<<grid, 256, 0, stream>>>(x, W, b, lora_A, lora_B, out);
}